// GeneralizedSSDLossNew_4389456576978
// MI455X (gfx1250) — compile-verified
//
#include <hip/hip_runtime.h>
#include <stdint.h>

#define BB 32
#define PP 32768
#define CC 81
#define NT 8   // tiles (of 32 priors) per wave, double-buffered

typedef unsigned int v4u __attribute__((ext_vector_type(4)));
typedef int         v8i __attribute__((ext_vector_type(8)));
typedef int         v4i __attribute__((ext_vector_type(4)));

__device__ __forceinline__ float smoothl1(float d) {
    float ad = fabsf(d);
    return ad < 1.f ? 0.5f * d * d : ad - 0.5f;
}
// order-preserving float->uint (ascending), so top-k = largest u
__device__ __forceinline__ unsigned f2ord(float f) {
    unsigned u = __float_as_uint(f);
    return (u & 0x80000000u) ? ~u : (u | 0x80000000u);
}
__device__ __forceinline__ float ord2f(unsigned u) {
    unsigned b = (u & 0x80000000u) ? (u & 0x7FFFFFFFu) : ~u;
    return __uint_as_float(b);
}

// ---------------- kernel 0: zero accumulators in workspace ----------------
__global__ void init_kernel(int* numpos, float* pos_ce, float* accs) {
    int i = threadIdx.x;
    if (i < BB) { numpos[i] = 0; pos_ce[i] = 0.f; }
    if (i < 2)  { accs[i] = 0.f; }
}

// ---------------- kernel 1: CE + smooth-L1, double-buffered TDM pipeline ---
__global__ __launch_bounds__(64) void ce_kernel(
    const float* __restrict__ loc_data, const float* __restrict__ conf_data,
    const float* __restrict__ loc_t,    const int* __restrict__ conf_t,
    float* __restrict__ mine, int* __restrict__ numpos,
    float* __restrict__ pos_ce, float* __restrict__ accs)
{
    __shared__ __align__(128) float tile[2][2][32 * CC];   // [wave][slab], 41472 B
    const int wave = threadIdx.x >> 5;
    const int lane = threadIdx.x & 31;
    // each wave owns NT consecutive 32-prior tiles = 256 contiguous priors;
    // P % 256 == 0, so a wave never crosses a batch row -> b is wave-uniform
    const long long t0    = ((long long)blockIdx.x * 2 + wave) * NT;
    const long long base0 = t0 << 5;
    const int b = (int)(base0 >> 15);

    // constant part of the TDM descriptor (2D tensor [B*P, 81] f32, tile 32x81)
    v8i g1;
    g1[0] = 0x00020000;            // data_size=2 (4 bytes), no multicast/pad/iterate
    g1[1] = (CC << 16);            // tensor_dim0=81 (low 16 of bits 79:48)
    g1[2] = 0;                     // dim0 hi16 = 0, tensor_dim1 lo16 = 0
    g1[3] = (CC << 16) | 0x10;     // tile_dim0=81 | tensor_dim1 hi16 (B*P=0x100000)
    g1[4] = 32;                    // tile_dim1=32 rows
    g1[5] = CC;                    // tensor_dim0_stride=81 (lo32)
    g1[6] = 0;
    g1[7] = 0;
    v4i gz4 = {0, 0, 0, 0};
    v8i gz8 = {0, 0, 0, 0, 0, 0, 0, 0};

    auto issue = [&](int i, int slab) {
        unsigned lds_addr = (unsigned)(uintptr_t)(&tile[wave][slab][0]);
        unsigned long long ga = (unsigned long long)(uintptr_t)conf_data +
                                (unsigned long long)(base0 + (long long)i * 32) * (CC * 4);
        v4u g0;
        g0[0] = 1u;                                     // count=1 valid descriptor
        g0[1] = lds_addr;                               // lds_addr [63:32]
        g0[2] = (unsigned)(ga & 0xFFFFFFFFull);         // global_addr lo
        g0[3] = ((unsigned)(ga >> 32) & 0x01FFFFFFu)    // global_addr hi (57-bit)
                | 0x80000000u;                          // type=2 [127:126]
        __builtin_amdgcn_tensor_load_to_lds(g0, g1, gz4, gz4, gz8, 0);
    };

    issue(0, 0);                   // prime the pipeline

    float ll_acc = 0.f, pce_acc = 0.f;
    int cnt_acc = 0;

    for (int i = 0; i < NT; ++i) {
        const long long gi = base0 + (long long)i * 32 + lane;
        // independent global loads overlap the in-flight TDM transfer
        int   lbl = conf_t[gi];
        float4 ld = ((const float4*)loc_data)[gi];
        float4 lt = ((const float4*)loc_t)[gi];

        if (i + 1 < NT) {
            // slab (i+1)&1 was consumed at iteration i-1; drain its ds reads
            asm volatile("s_wait_dscnt 0x0" ::: "memory");
            issue(i + 1, (i + 1) & 1);
            __builtin_amdgcn_s_wait_tensorcnt(1);  // in-order: tile i complete
        } else {
            __builtin_amdgcn_s_wait_tensorcnt(0);
        }
        asm volatile("" ::: "memory");

        // logsumexp over this lane's 81-float LDS row (stride 81 -> conflict-free)
        const float* r = &tile[wave][i & 1][lane * CC];
        float mx = r[0];
        #pragma unroll
        for (int c = 1; c < CC; ++c) mx = fmaxf(mx, r[c]);
        float s = 0.f;
        #pragma unroll
        for (int c = 0; c < CC; ++c) s += __expf(r[c] - mx);
        float lse = mx + __logf(s);

        float ce  = lse - r[lbl];
        bool  pos = lbl > 0;
        mine[gi] = pos ? 0.f : ce;   // loss_mine (positives zeroed) for mining

        float ll = smoothl1(ld.x - lt.x) + smoothl1(ld.y - lt.y) +
                   smoothl1(ld.z - lt.z) + smoothl1(ld.w - lt.w);
        ll_acc  += pos ? ll : 0.f;
        pce_acc += pos ? ce : 0.f;
        cnt_acc += pos ? 1 : 0;
    }

    // one wave32 reduction + 3 atomics per 256 priors
    #pragma unroll
    for (int m = 16; m > 0; m >>= 1) {
        ll_acc  += __shfl_xor(ll_acc,  m, 32);
        pce_acc += __shfl_xor(pce_acc, m, 32);
        cnt_acc += __shfl_xor(cnt_acc, m, 32);
    }
    if (lane == 0) {
        atomicAdd(&accs[0], ll_acc);
        atomicAdd(&pos_ce[b], pce_acc);
        atomicAdd(&numpos[b], cnt_acc);
    }
}

// ---------------- kernel 2: per-row radix-select top-k + conf loss ---------
__global__ __launch_bounds__(1024) void select_kernel(
    const float* __restrict__ mine, const int* __restrict__ numpos,
    const float* __restrict__ pos_ce, float* __restrict__ accs)
{
    const int b = blockIdx.x;
    const float* row = mine + (size_t)b * PP;
    __shared__ unsigned hist[256];
    __shared__ unsigned s_prefix, s_kk, s_cnt;
    __shared__ float s_sum;

    int np      = numpos[b];
    int num_neg = min(3 * np, PP - 1);
    int M       = PP - np;                 // negatives all have ce>0 > 0(=positives)
    int k       = min(num_neg, M);         // exact count of selected negatives

    if (threadIdx.x == 0) { s_prefix = 0u; s_kk = (unsigned)k; s_sum = 0.f; s_cnt = 0u; }
    __syncthreads();

    if (k > 0) {
        // 4-pass MSB-first radix select: find k-th largest ordered value
        for (int shift = 24; shift >= 0; shift -= 8) {
            for (int i = threadIdx.x; i < 256; i += blockDim.x) hist[i] = 0u;
            __syncthreads();
            unsigned pfx     = s_prefix;
            unsigned mask_hi = (shift == 24) ? 0u : (0xFFFFFFFFu << (shift + 8));
            for (int i = threadIdx.x; i < PP; i += blockDim.x) {
                unsigned u = f2ord(row[i]);
                if ((u & mask_hi) == (pfx & mask_hi))
                    atomicAdd(&hist[(u >> shift) & 0xFFu], 1u);
            }
            __syncthreads();
            if (threadIdx.x == 0) {
                unsigned kk = s_kk, cum = 0u, digit = 0u;
                for (int d = 255; d >= 0; --d) {
                    unsigned c = hist[d];
                    if (cum + c >= kk) { digit = (unsigned)d; s_kk = kk - cum; break; }
                    cum += c;
                }
                s_prefix = pfx | (digit << shift);
            }
            __syncthreads();
        }
        // sum all strictly-above-threshold, fix up ties by value
        unsigned T = s_prefix;
        float sum_gt = 0.f; unsigned cnt_gt = 0u;
        for (int i = threadIdx.x; i < PP; i += blockDim.x) {
            float v = row[i];
            if (f2ord(v) > T) { sum_gt += v; cnt_gt++; }
        }
        #pragma unroll
        for (int m = 16; m > 0; m >>= 1) {
            sum_gt += __shfl_xor(sum_gt, m, 32);
            cnt_gt += __shfl_xor(cnt_gt, m, 32);
        }
        if ((threadIdx.x & 31) == 0) {
            atomicAdd(&s_sum, sum_gt);
            atomicAdd(&s_cnt, cnt_gt);
        }
        __syncthreads();
    }
    if (threadIdx.x == 0) {
        float row_loss = pos_ce[b];
        if (k > 0)
            row_loss += s_sum + (float)((unsigned)k - s_cnt) * ord2f(s_prefix);
        atomicAdd(&accs[1], row_loss);
    }
}

// ---------------- kernel 3: finalize -------------------------------------
__global__ void finalize_kernel(const int* __restrict__ numpos,
                                const float* __restrict__ accs,
                                float* __restrict__ out)
{
    float N = 0.f;
    for (int i = 0; i < BB; ++i) N += (float)numpos[i];
    out[0] = accs[0] / N;   // loss_l / N * LOC_WEIGHT(=1)
    out[1] = accs[1] / N;   // loss_c / N * CONF_WEIGHT(=1)
}

extern "C" void kernel_launch(void* const* d_in, const int* in_sizes, int n_in,
                              void* d_out, int out_size, void* d_ws, size_t ws_size,
                              hipStream_t stream) {
    const float* loc_data  = (const float*)d_in[0];
    const float* conf_data = (const float*)d_in[1];
    const float* loc_t     = (const float*)d_in[2];
    const int*   conf_t    = (const int*)d_in[3];
    float* out = (float*)d_out;

    // workspace layout: mine[B*P] f32 | numpos[B] i32 | pos_ce[B] f32 | accs[2] f32
    float* ws_mine   = (float*)d_ws;
    int*   ws_numpos = (int*)(ws_mine + (size_t)BB * PP);
    float* ws_pos_ce = (float*)(ws_numpos + BB);
    float* ws_accs   = ws_pos_ce + BB;

    init_kernel<<<1, 64, 0, stream>>>(ws_numpos, ws_pos_ce, ws_accs);
    // 2 waves/block * NT tiles * 32 priors = 512 priors per block
    ce_kernel<<<(BB * PP) / (2 * NT * 32), 64, 0, stream>>>(
        loc_data, conf_data, loc_t, conf_t, ws_mine, ws_numpos, ws_pos_ce, ws_accs);
    select_kernel<<<BB, 1024, 0, stream>>>(ws_mine, ws_numpos, ws_pos_ce, ws_accs);
    finalize_kernel<<<1, 1, 0, stream>>>(ws_numpos, ws_accs, out);
}